// CrossAttentionModule_23776938951334
// MI455X (gfx1250) — compile-verified
//
#include <hip/hip_runtime.h>
#include <hip/hip_bf16.h>

// Problem constants (from reference): B=1, C=64, D=64, H=64, W=8
#define NC   64
#define ND   64
#define NH   64
#define NW   8
#define NSP  (ND * NH * NW)     // 32768 voxels
#define NHEADS 8
#define HDIM   8
#define K3     27

typedef float v2f __attribute__((ext_vector_type(2)));
typedef float v8f __attribute__((ext_vector_type(8)));

// ---------------------------------------------------------------------------
// Kernel 1: per-channel instance-norm statistics for skip (t=0) and up (t=1).
// stats layout (floats): [0..63] mean_skip, [64..127] invstd_skip,
//                        [128..191] mean_up, [192..255] invstd_up
// ---------------------------------------------------------------------------
__global__ __launch_bounds__(256) void ca_stats_kernel(
    const float* __restrict__ skip, const float* __restrict__ up,
    float* __restrict__ stats) {
  const int bx = blockIdx.x;          // 0..127
  const int t = bx >> 6;              // tensor select
  const int c = bx & 63;              // channel
  const float* src = (t == 0 ? skip : up) + (size_t)c * NSP;

  float s1 = 0.f, s2 = 0.f;
  for (int i = threadIdx.x; i < NSP; i += 256) {
    float x = src[i];
    s1 += x;
    s2 += x * x;
  }
  __shared__ float r1[256];
  __shared__ float r2[256];
  r1[threadIdx.x] = s1;
  r2[threadIdx.x] = s2;
  __syncthreads();
  for (int off = 128; off > 0; off >>= 1) {
    if ((int)threadIdx.x < off) {
      r1[threadIdx.x] += r1[threadIdx.x + off];
      r2[threadIdx.x] += r2[threadIdx.x + off];
    }
    __syncthreads();
  }
  if (threadIdx.x == 0) {
    const float invN = 1.0f / (float)NSP;
    float mean = r1[0] * invN;
    float var  = r2[0] * invN - mean * mean;
    stats[t * 128 + c]      = mean;
    stats[t * 128 + 64 + c] = rsqrtf(var + 1e-5f);
  }
}

// ---------------------------------------------------------------------------
// Kernel 2: fused normalize + QK projection (skip_n @ w_qk + b_qk, N=128)
//           and V projection (up_n @ w_v + b_v, N=64), via fp32 WMMA.
// One wave handles one 16-row spatial tile (M=16, K=64 as 16 steps of K=4).
// qkb: [NSP][128] row-major (cols 0..63 = q, 64..127 = k)
// vb : [NSP][64]  row-major
// ---------------------------------------------------------------------------
__global__ __launch_bounds__(256) void ca_qkv_kernel(
    const float* __restrict__ skip, const float* __restrict__ up,
    const float* __restrict__ w_qk, const float* __restrict__ b_qk,
    const float* __restrict__ w_v,  const float* __restrict__ b_v,
    const float* __restrict__ stats,
    float* __restrict__ qkb, float* __restrict__ vb) {
  const int lane = threadIdx.x & 31;
  const int wave = threadIdx.x >> 5;
  const int tile = blockIdx.x * 8 + wave;      // 0..2047
  const int m    = lane & 15;                  // A-row / D-col index
  const int kh   = lane >> 4;                  // K half-select (0 or 1)
  const int row0 = tile * 16;
  const int row  = row0 + m;

  // Cache all 16 K-step A fragments (normalized skip and up) in registers.
  v2f aS[16], aU[16];
#pragma unroll
  for (int ks = 0; ks < 16; ++ks) {
    const int c0 = ks * 4 + kh * 2;
    aS[ks].x = (skip[(size_t)c0 * NSP + row]       - stats[c0])         * stats[64 + c0];
    aS[ks].y = (skip[(size_t)(c0 + 1) * NSP + row] - stats[c0 + 1])     * stats[64 + c0 + 1];
    aU[ks].x = (up  [(size_t)c0 * NSP + row]       - stats[128 + c0])   * stats[192 + c0];
    aU[ks].y = (up  [(size_t)(c0 + 1) * NSP + row] - stats[128 + c0 + 1]) * stats[192 + c0 + 1];
  }

  // ---- QK projection: 8 N-tiles of 16 columns, K = 64 ----
  for (int nt = 0; nt < 8; ++nt) {
    v8f acc = 0.f;
    const int n = nt * 16 + m;
#pragma unroll
    for (int ks = 0; ks < 16; ++ks) {
      const int c0 = ks * 4 + kh * 2;
      v2f b;
      b.x = w_qk[c0 * 128 + n];
      b.y = w_qk[(c0 + 1) * 128 + n];
      acc = __builtin_amdgcn_wmma_f32_16x16x4_f32(
          false, aS[ks], false, b, (short)0, acc, false, false);
    }
    const float bias = b_qk[n];
#pragma unroll
    for (int r = 0; r < 8; ++r) {
      const int mm = r + kh * 8;
      qkb[(size_t)(row0 + mm) * 128 + n] = acc[r] + bias;
    }
  }

  // ---- V projection: 4 N-tiles of 16 columns, K = 64 ----
  for (int nt = 0; nt < 4; ++nt) {
    v8f acc = 0.f;
    const int n = nt * 16 + m;
#pragma unroll
    for (int ks = 0; ks < 16; ++ks) {
      const int c0 = ks * 4 + kh * 2;
      v2f b;
      b.x = w_v[c0 * 64 + n];
      b.y = w_v[(c0 + 1) * 64 + n];
      acc = __builtin_amdgcn_wmma_f32_16x16x4_f32(
          false, aU[ks], false, b, (short)0, acc, false, false);
    }
    const float bias = b_v[n];
#pragma unroll
    for (int r = 0; r < 8; ++r) {
      const int mm = r + kh * 8;
      vb[(size_t)(row0 + mm) * 64 + n] = acc[r] + bias;
    }
  }
}

// ---------------------------------------------------------------------------
// Kernel 3: 3x3x3 neighborhood attention + RPB + softmax.
// One thread per (voxel, head). Writes ao[voxel][64] (head-major channels).
// ---------------------------------------------------------------------------
__global__ __launch_bounds__(256) void ca_attn_kernel(
    const float* __restrict__ qkb, const float* __restrict__ vb,
    const float* __restrict__ rpb, float* __restrict__ ao) {
  const int gid   = blockIdx.x * 256 + threadIdx.x;   // 0..262143
  const int h     = gid & 7;
  const int voxel = gid >> 3;
  const int w  = voxel & 7;
  const int hh = (voxel >> 3) & 63;
  const int d  = voxel >> 9;

  // Clipped window starts: clip(i-1, 0, L-3)
  int sd = d - 1;  sd = sd < 0 ? 0 : (sd > ND - 3 ? ND - 3 : sd);
  int sh = hh - 1; sh = sh < 0 ? 0 : (sh > NH - 3 ? NH - 3 : sh);
  int sw = w - 1;  sw = sw < 0 ? 0 : (sw > NW - 3 ? NW - 3 : sw);

  float q[HDIM];
#pragma unroll
  for (int t = 0; t < HDIM; ++t) q[t] = qkb[(size_t)voxel * 128 + h * 8 + t];

  const float scale = 0.35355339059327373f;  // 1/sqrt(8)
  float logits[K3];
  int   nvox[K3];

#pragma unroll
  for (int jd = 0; jd < 3; ++jd) {
#pragma unroll
    for (int jh = 0; jh < 3; ++jh) {
#pragma unroll
      for (int jw = 0; jw < 3; ++jw) {
        const int j  = (jd * 3 + jh) * 3 + jw;
        const int nv = (sd + jd) * (NH * NW) + (sh + jh) * NW + (sw + jw);
        const float* kp = qkb + (size_t)nv * 128 + 64 + h * 8;
        float dot = 0.f;
#pragma unroll
        for (int t = 0; t < HDIM; ++t) dot += q[t] * kp[t];
        const int rd = sd + jd - d + 2;
        const int rh = sh + jh - hh + 2;
        const int rw = sw + jw - w + 2;
        logits[j] = dot * scale + rpb[((h * 5 + rd) * 5 + rh) * 5 + rw];
        nvox[j] = nv;
      }
    }
  }

  float mx = logits[0];
#pragma unroll
  for (int i = 1; i < K3; ++i) mx = fmaxf(mx, logits[i]);

  float acc[HDIM];
#pragma unroll
  for (int t = 0; t < HDIM; ++t) acc[t] = 0.f;
  float sum = 0.f;
#pragma unroll
  for (int i = 0; i < K3; ++i) {
    const float p = __expf(logits[i] - mx);
    sum += p;
    const float* vp = vb + (size_t)nvox[i] * 64 + h * 8;
#pragma unroll
    for (int t = 0; t < HDIM; ++t) acc[t] += p * vp[t];
  }
  const float inv = 1.f / sum;
#pragma unroll
  for (int t = 0; t < HDIM; ++t)
    ao[(size_t)voxel * 64 + h * 8 + t] = acc[t] * inv;
}

// ---------------------------------------------------------------------------
// Kernel 4: output projection (ao @ w_proj + b_proj), stored channel-major
// out[c][spatial] to match the reference's final transpose. fp32 WMMA.
// ---------------------------------------------------------------------------
__global__ __launch_bounds__(256) void ca_proj_kernel(
    const float* __restrict__ ao, const float* __restrict__ w_proj,
    const float* __restrict__ b_proj, float* __restrict__ out) {
  const int lane = threadIdx.x & 31;
  const int wave = threadIdx.x >> 5;
  const int tile = blockIdx.x * 8 + wave;      // 0..2047
  const int m    = lane & 15;
  const int kh   = lane >> 4;
  const int row0 = tile * 16;
  const int row  = row0 + m;

  v2f a[16];
#pragma unroll
  for (int ks = 0; ks < 16; ++ks) {
    const int c0 = ks * 4 + kh * 2;
    a[ks].x = ao[(size_t)row * 64 + c0];
    a[ks].y = ao[(size_t)row * 64 + c0 + 1];
  }

  for (int nt = 0; nt < 4; ++nt) {
    v8f acc = 0.f;
    const int n = nt * 16 + m;
#pragma unroll
    for (int ks = 0; ks < 16; ++ks) {
      const int c0 = ks * 4 + kh * 2;
      v2f b;
      b.x = w_proj[c0 * 64 + n];
      b.y = w_proj[(c0 + 1) * 64 + n];
      acc = __builtin_amdgcn_wmma_f32_16x16x4_f32(
          false, a[ks], false, b, (short)0, acc, false, false);
    }
    const float bias = b_proj[n];
#pragma unroll
    for (int r = 0; r < 8; ++r) {
      const int mm = r + kh * 8;
      out[(size_t)n * NSP + row0 + mm] = acc[r] + bias;  // out[C][D][H][W]
    }
  }
}

// ---------------------------------------------------------------------------
// Host launcher
// ---------------------------------------------------------------------------
extern "C" void kernel_launch(void* const* d_in, const int* in_sizes, int n_in,
                              void* d_out, int out_size, void* d_ws, size_t ws_size,
                              hipStream_t stream) {
  const float* skip   = (const float*)d_in[0];
  const float* up     = (const float*)d_in[1];
  const float* w_qk   = (const float*)d_in[2];
  const float* b_qk   = (const float*)d_in[3];
  const float* w_v    = (const float*)d_in[4];
  const float* b_v    = (const float*)d_in[5];
  const float* w_proj = (const float*)d_in[6];
  const float* b_proj = (const float*)d_in[7];
  const float* rpb    = (const float*)d_in[8];
  float* out = (float*)d_out;

  // Workspace layout (floats):
  float* ws    = (float*)d_ws;
  float* stats = ws;                                 // 256 floats
  float* qkb   = ws + 256;                           // NSP * 128
  float* vb    = qkb + (size_t)NSP * 128;            // NSP * 64
  float* ao    = vb + (size_t)NSP * 64;              // NSP * 64

  ca_stats_kernel<<<128, 256, 0, stream>>>(skip, up, stats);
  ca_qkv_kernel<<<NSP / 16 / 8, 256, 0, stream>>>(skip, up, w_qk, b_qk, w_v, b_v,
                                                  stats, qkb, vb);
  ca_attn_kernel<<<(NSP * NHEADS) / 256, 256, 0, stream>>>(qkb, vb, rpb, ao);
  ca_proj_kernel<<<NSP / 16 / 8, 256, 0, stream>>>(ao, w_proj, b_proj, out);
}